// DepthCueExtractor_77481210020173
// MI455X (gfx1250) — compile-verified
//
#include <hip/hip_runtime.h>
#include <hip/hip_bf16.h>

// Problem sizes (fixed by the reference)
#define BB 4
#define UU 81
#define HH 64
#define WW 64
#define CC 4
#define FF 32

typedef __attribute__((ext_vector_type(2))) float v2f;
typedef __attribute__((ext_vector_type(8))) float v8f;

// ---------------------------------------------------------------------------
// Kernel 1: mask[b][y][f] = mean_h(f_maps[b,h,y,f]) / max_w(mean_h(...))
// f_maps layout: [B, H, W, F] (F innermost). mask stored [B][H][F] so the
// main kernel can read 16 contiguous f values per half-wave.
// ---------------------------------------------------------------------------
__global__ void dce_mask_kernel(const float* __restrict__ fmaps,
                                float* __restrict__ mask) {
    __shared__ float hv[WW * FF];   // 2048 floats = 8 KB
    __shared__ float mxf[FF];

    const int b = blockIdx.x;
    const int t = threadIdx.x;      // 256 threads
    const float* fb = fmaps + (size_t)b * (HH * WW * FF);

    // mean over h for each (w,f) pair; p = w*32 + f, contiguous across threads
    #pragma unroll
    for (int i = 0; i < 8; ++i) {
        const int p = t + i * 256;
        float s = 0.0f;
        for (int h = 0; h < HH; ++h) {
            s += fb[(size_t)h * (WW * FF) + p];
        }
        hv[p] = s * (1.0f / 64.0f);
    }
    __syncthreads();

    // max over w per f
    if (t < FF) {
        float mx = hv[t];
        for (int w = 1; w < WW; ++w) mx = fmaxf(mx, hv[w * FF + t]);
        mxf[t] = mx;
    }
    __syncthreads();

    // normalized mask (note: h_mask[b,f,y] == hv[b, w=y, f] / mx[b,f])
    #pragma unroll
    for (int i = 0; i < 8; ++i) {
        const int p = t + i * 256;
        mask[(size_t)b * (HH * FF) + p] = hv[p] / mxf[p & (FF - 1)];
    }
}

// ---------------------------------------------------------------------------
// Kernel 2: out[b,u,y,x,f] = mean_c(lfi[b,u,y,x,:]) * mask[b,y,f]
// Per (b,u,y) row this is a rank-1 outer product xm(64) (x) mrow(32),
// computed as 4x2 tiles of V_WMMA_F32_16X16X4_F32 with only K=0 populated
// (exact fp32 multiply, RNE — identical to scalar v_mul_f32).
//
// Grid: B*U*4 blocks. Block: 256 threads = 8 waves. Each block owns a
// 16-row y-quarter of one (b,u); each wave handles 2 rows.
// ---------------------------------------------------------------------------
__global__ void dce_main_kernel(const float* __restrict__ lfi,
                                const float* __restrict__ mask,
                                float* __restrict__ out) {
    const int bu   = blockIdx.x >> 2;      // 0..B*U-1
    const int yq   = blockIdx.x & 3;       // y quarter
    const int wave = threadIdx.x >> 5;     // 0..7
    const int lane = threadIdx.x & 31;
    const int b    = bu / UU;

    const float* lrow_base = lfi  + (size_t)bu * (HH * WW * CC);
    float*       orow_base = out  + (size_t)bu * (HH * WW * FF);
    const float* mrow_base = mask + (size_t)b  * (HH * FF);

    const int  laneLo = lane & 15;
    const bool lo     = lane < 16;
    const int  xoff   = (lane >> 4) * 8;   // D-layout: lanes 16-31 hold M+8

    for (int ry = 0; ry < 2; ++ry) {
        const int y = yq * 16 + wave * 2 + ry;
        const float* lrow = lrow_base + (size_t)y * (WW * CC);
        float*       orow = orow_base + (size_t)y * (WW * FF);
        const float* mrow = mrow_base + y * FF;

        // B operands: K=0 row of 4x16 B-matrix lives in lanes 0-15 of VGPR0
        v2f bop0 = {0.0f, 0.0f};
        v2f bop1 = {0.0f, 0.0f};
        if (lo) {
            bop0.x = mrow[laneLo];          // f = 0..15
            bop1.x = mrow[16 + laneLo];     // f = 16..31
        }

        #pragma unroll
        for (int xt = 0; xt < 4; ++xt) {
            // A operand: K=0 column of 16x4 A-matrix in lanes 0-15 of VGPR0
            v2f aop = {0.0f, 0.0f};
            if (lo) {
                const float* p = lrow + (size_t)(xt * 16 + laneLo) * CC;
                aop.x = (p[0] + p[1] + p[2] + p[3]) * 0.25f;  // mean over c
            }

            v8f d0 = {};
            v8f d1 = {};
            d0 = __builtin_amdgcn_wmma_f32_16x16x4_f32(
                     false, aop, false, bop0, (short)0, d0, false, false);
            d1 = __builtin_amdgcn_wmma_f32_16x16x4_f32(
                     false, aop, false, bop1, (short)0, d1, false, false);

            // D layout: VGPR r, lanes 0-15 -> (M=r, N=lane); lanes 16-31 ->
            // (M=r+8, N=lane-16). Here M = x within tile, N = f within tile.
            float* o = orow + (size_t)(xt * 16 + xoff) * FF + laneLo;
            #pragma unroll
            for (int r = 0; r < 8; ++r) {
                o[r * FF]      = d0[r];   // f tile 0
                o[r * FF + 16] = d1[r];   // f tile 1
            }
        }
    }
}

// ---------------------------------------------------------------------------
extern "C" void kernel_launch(void* const* d_in, const int* in_sizes, int n_in,
                              void* d_out, int out_size, void* d_ws, size_t ws_size,
                              hipStream_t stream) {
    const float* lfi   = (const float*)d_in[0];  // [B,U,H,W,C]
    const float* fmaps = (const float*)d_in[1];  // [B,H,W,F]
    float* out  = (float*)d_out;                 // [B,U,H,W,F]
    float* mask = (float*)d_ws;                  // [B,H,F] scratch (32 KB)

    dce_mask_kernel<<<BB, 256, 0, stream>>>(fmaps, mask);
    dce_main_kernel<<<BB * UU * 4, 256, 0, stream>>>(lfi, mask, out);
}